// SelfAttention_82274393522993
// MI455X (gfx1250) — compile-verified
//
#include <hip/hip_runtime.h>
#include <math.h>

typedef float v2f __attribute__((ext_vector_type(2)));
typedef float v8f __attribute__((ext_vector_type(8)));
typedef int   v4i_async __attribute__((vector_size(16)));   // matches builtin param type

#define NEG_MASK_VAL (-1000000000.0f)

constexpr int B = 8, S = 16, T = 128, D = 64;
constexpr int N = S * T;           // 2048
constexpr int NROWS = B * N;       // 16384
constexpr int LDP = 2052;          // padded LDS row stride (floats); 2052%64==4 -> conflict-free
constexpr int LDS_P_FLOATS = 16 * LDP;     // 32832 floats = 131,328 B
constexpr int LDS_S_FLOATS = 8 * 256;      // partial 16x16 tiles for K-split reduce
constexpr size_t FUSED_LDS_BYTES = (size_t)(LDS_P_FLOATS + LDS_S_FLOATS) * sizeof(float);

#if defined(__has_builtin)
#  if __has_builtin(__builtin_amdgcn_global_store_async_from_lds_b128)
#    define HAVE_ASYNC_STORE 1
#  else
#    define HAVE_ASYNC_STORE 0
#  endif
#else
#  define HAVE_ASYNC_STORE 0
#endif

__device__ inline v8f wmma_f32_step(v2f a, v2f b, v8f c) {
  // D = A(16x4 f32) * B(4x16 f32) + C(16x16 f32)
  return __builtin_amdgcn_wmma_f32_16x16x4_f32(
      /*neg_a=*/false, a, /*neg_b=*/false, b,
      /*c_mod=*/(short)0, c, /*reuse_a=*/false, /*reuse_b=*/false);
}

// ---------------------------------------------------------------------------
// Kernel 1: q/k/v projections. 4 waves/block, each wave owns 16 output cols.
// A-fragments (x tile) hoisted and reused for the 3 GEMMs.
// v stored transposed: vf[b, t*S + s, :]  (reference quirk).
// ---------------------------------------------------------------------------
__global__ __launch_bounds__(128) void qkv_wmma_kernel(
    const float* __restrict__ x,
    const float* __restrict__ Wq, const float* __restrict__ bq,
    const float* __restrict__ Wk, const float* __restrict__ bk,
    const float* __restrict__ Wv, const float* __restrict__ bv,
    float* __restrict__ q, float* __restrict__ k, float* __restrict__ vf)
{
  const int wave = threadIdx.x >> 5;
  const int lane = threadIdx.x & 31;
  const int half = lane >> 4;
  const int l16  = lane & 15;
  const int row0 = blockIdx.x * 16;
  const int j0   = wave * 16;

  v2f afrag[16];
  #pragma unroll
  for (int kk = 0; kk < 16; ++kk) {
    const float* ap = x + (size_t)(row0 + l16) * D + kk * 4 + 2 * half;
    afrag[kk].x = ap[0];
    afrag[kk].y = ap[1];
  }

  const float* Ws[3] = {Wq, Wk, Wv};
  const float* bs[3] = {bq, bk, bv};

  #pragma unroll
  for (int m = 0; m < 3; ++m) {
    const float* W = Ws[m];
    v8f c = {};
    #pragma unroll
    for (int kk = 0; kk < 16; ++kk) {
      v2f bfrag;  // B[kd][jc] = W[(j0+jc)*D + kd]  (x @ W.T)
      const float* bp = W + (size_t)(j0 + l16) * D + kk * 4 + 2 * half;
      bfrag.x = bp[0];
      bfrag.y = bp[1];
      c = wmma_f32_step(afrag[kk], bfrag, c);
    }
    const float bias = bs[m][j0 + l16];
    #pragma unroll
    for (int r = 0; r < 8; ++r) {
      const int g = row0 + r + 8 * half;
      const float val = c[r] + bias;
      if (m == 0) {
        q[(size_t)g * D + j0 + l16] = val;
      } else if (m == 1) {
        k[(size_t)g * D + j0 + l16] = val;
      } else {
        const int bidx = g >> 11;
        const int rem  = g & (N - 1);
        const int s = rem >> 7, t = rem & (T - 1);
        vf[((size_t)bidx * N + t * S + s) * D + j0 + l16] = val;
      }
    }
  }
}

// ---------------------------------------------------------------------------
// Kernel 2 (fused): for one 16-row block of one batch:
//   Phase 1: masked/scaled scores (16 x 2048) -> LDS  (WMMA QK^T)
//   Phase 2: softmax rows in LDS; emit p_attn once (async LDS->global)
//   Phase 3: ctx tile (16 x 64) = P_lds @ Vf         (WMMA, K split 2 ways)
// 8 waves / block. LDS: 16*2052 f32 P-buffer + 8*256 f32 partial scratch.
// ---------------------------------------------------------------------------
__global__ __launch_bounds__(256) void fused_attn_kernel(
    const float* __restrict__ q, const float* __restrict__ k,
    const float* __restrict__ vf,
    const int* __restrict__ tags, const float* __restrict__ basemask,
    float* __restrict__ p_out, float* __restrict__ ctx)
{
  extern __shared__ float lds[];
  float* ldsP = lds;                       // 16 x LDP
  float* ldsS = lds + LDS_P_FLOATS;        // 8 x 256

  const int wave = threadIdx.x >> 5;
  const int lane = threadIdx.x & 31;
  const int half = lane >> 4;
  const int l16  = lane & 15;
  const int b    = blockIdx.y;
  const int n0   = blockIdx.x * 16;

  const float* qb = q + (size_t)b * N * D;
  const float* kb = k + (size_t)b * N * D;

  // ---------- Phase 1: scores row-block into LDS ----------
  v2f afrag[16];
  #pragma unroll
  for (int kk = 0; kk < 16; ++kk) {
    const float* ap = qb + (size_t)(n0 + l16) * D + kk * 4 + 2 * half;
    afrag[kk].x = ap[0];
    afrag[kk].y = ap[1];
  }

  for (int mt = wave; mt < N / 16; mt += 8) {
    const int m0 = mt * 16;
    v8f c = {};
    #pragma unroll
    for (int kk = 0; kk < 16; ++kk) {
      v2f bfrag;  // B[kd][mc] = K[(m0+mc)*D + kd]   (K^T)
      const float* bp = kb + (size_t)(m0 + l16) * D + kk * 4 + 2 * half;
      bfrag.x = bp[0];
      bfrag.y = bp[1];
      c = wmma_f32_step(afrag[kk], bfrag, c);
    }
    const int colm = m0 + l16;
    const int sm = colm >> 7, tm = colm & (T - 1);
    const int tag_m = tags[b * S + sm];
    #pragma unroll
    for (int r = 0; r < 8; ++r) {
      const int lrow = r + 8 * half;
      const int rown = n0 + lrow;
      const int sn = rown >> 7, tn = rown & (T - 1);
      const bool ok = (tags[b * S + sn] == tag_m) && (basemask[tn * T + tm] != 0.0f);
      ldsP[lrow * LDP + colm] = ok ? c[r] * 0.125f : NEG_MASK_VAL;
    }
  }
  __syncthreads();

  // ---------- Phase 2: softmax (2 rows per wave), write p_attn ----------
  float* pgb = p_out + (size_t)b * N * N;
  #pragma unroll
  for (int rr = 0; rr < 2; ++rr) {
    const int lrow = wave * 2 + rr;
    float* rowp = ldsP + lrow * LDP;
    float vreg[64];
    float mx = -INFINITY;
    #pragma unroll
    for (int i = 0; i < 64; ++i) {
      vreg[i] = rowp[lane + 32 * i];
      mx = fmaxf(mx, vreg[i]);
    }
    #pragma unroll
    for (int off = 16; off >= 1; off >>= 1) mx = fmaxf(mx, __shfl_xor(mx, off, 32));
    float sum = 0.0f;
    #pragma unroll
    for (int i = 0; i < 64; ++i) {
      vreg[i] = __expf(vreg[i] - mx);
      sum += vreg[i];
    }
    #pragma unroll
    for (int off = 16; off >= 1; off >>= 1) sum += __shfl_xor(sum, off, 32);
    const float inv = 1.0f / sum;
    float* grow = pgb + (size_t)(n0 + lrow) * N;
    #pragma unroll
    for (int i = 0; i < 64; ++i) {
      const float pv = vreg[i] * inv;
      rowp[lane + 32 * i] = pv;
#if !HAVE_ASYNC_STORE
      grow[lane + 32 * i] = pv;
#endif
    }
#if HAVE_ASYNC_STORE
    // make LDS writes visible to the async engine, then stream row out
    asm volatile("s_wait_dscnt 0x0" ::: "memory");
    #pragma unroll
    for (int i = 0; i < 16; ++i) {
      __builtin_amdgcn_global_store_async_from_lds_b128(
          (v4i_async*)(grow + lane * 4 + i * 128),
          (v4i_async*)(rowp + lane * 4 + i * 128),
          0, 0);
    }
#endif
  }
  __syncthreads();

  // ---------- Phase 3: ctx = P_lds(16x2048) @ Vf(2048x64) ----------
  const float* vb = vf + (size_t)b * N * D;
  const int jt    = wave & 3;       // which 16-col tile of D
  const int j0    = jt * 16;
  const int khalf = wave >> 2;      // K split in halves of 1024
  v8f c = {};
  for (int kk = khalf * 256; kk < khalf * 256 + 256; ++kk) {
    v2f a, bfrag;
    const float* lp = ldsP + l16 * LDP + kk * 4 + 2 * half;
    a.x = lp[0];
    a.y = lp[1];
    const float* bp = vb + (size_t)(kk * 4 + 2 * half) * D + j0 + l16;
    bfrag.x = bp[0];
    bfrag.y = bp[D];
    c = wmma_f32_step(a, bfrag, c);
  }
  if (khalf == 1) {
    #pragma unroll
    for (int r = 0; r < 8; ++r)
      ldsS[(wave - 4) * 256 + (r + 8 * half) * 16 + l16] = c[r];
  }
  __syncthreads();
  if (khalf == 0) {
    #pragma unroll
    for (int r = 0; r < 8; ++r) {
      const float val = c[r] + ldsS[wave * 256 + (r + 8 * half) * 16 + l16];
      ctx[((size_t)b * N + n0 + r + 8 * half) * D + j0 + l16] = val;
    }
  }
#if HAVE_ASYNC_STORE
  asm volatile("s_wait_asynccnt 0x0" ::: "memory");
#endif
}

// ---------------------------------------------------------------------------
extern "C" void kernel_launch(void* const* d_in, const int* in_sizes, int n_in,
                              void* d_out, int out_size, void* d_ws, size_t ws_size,
                              hipStream_t stream) {
  const float* x    = (const float*)d_in[0];
  const int*   tags = (const int*)d_in[1];
  const float* mask = (const float*)d_in[2];
  const float* Wk_w = (const float*)d_in[3];
  const float* Wk_b = (const float*)d_in[4];
  const float* Wq_w = (const float*)d_in[5];
  const float* Wq_b = (const float*)d_in[6];
  const float* Wv_w = (const float*)d_in[7];
  const float* Wv_b = (const float*)d_in[8];

  float* ctx = (float*)d_out;                  // B*S*T*D
  float* p   = ctx + (size_t)B * N * D;        // B*N*N (p_attn, post-softmax)

  float* q  = (float*)d_ws;                    // 3 x 4 MB scratch
  float* k  = q + (size_t)NROWS * D;
  float* vf = k + (size_t)NROWS * D;

  qkv_wmma_kernel<<<dim3(NROWS / 16), dim3(128), 0, stream>>>(
      x, Wq_w, Wq_b, Wk_w, Wk_b, Wv_w, Wv_b, q, k, vf);

  fused_attn_kernel<<<dim3(N / 16, B), dim3(256), FUSED_LDS_BYTES, stream>>>(
      q, k, vf, tags, mask, p, ctx);
}